// GaussianLayer_2379411882180
// MI455X (gfx1250) — compile-verified
//
#include <hip/hip_runtime.h>

typedef __attribute__((ext_vector_type(2))) float v2f;
typedef __attribute__((ext_vector_type(8))) float v8f;

#define IMG   512
#define PADR  10
#define KS    21
#define TILE  64
#define HALO  84            // TILE + 2*PADR
#define INS   84            // LDS stride of input tile (floats)
#define TMPS  76            // LDS stride of intermediate tile (floats)

#ifndef __has_builtin
#define __has_builtin(x) 0
#endif
#if __has_builtin(__builtin_amdgcn_global_load_async_to_lds_b32) && \
    __has_builtin(__builtin_amdgcn_s_wait_asynccnt)
#define USE_ASYNC_LDS 1
#else
#define USE_ASYNC_LDS 0
#endif

typedef __attribute__((address_space(1))) int as1_int;
typedef __attribute__((address_space(3))) int as3_int;

__device__ __forceinline__ int reflect512(int p) {
    // ReflectionPad2d semantics (no edge repeat); PADR < IMG so one fold suffices
    p = (p < 0) ? -p : p;
    p = (p >= IMG) ? (2 * IMG - 2 - p) : p;
    return p;
}

__global__ __launch_bounds__(256)
void gauss21_sep_wmma(const float* __restrict__ x,
                      const float* __restrict__ W,
                      float* __restrict__ out) {
    __shared__ float s_in[HALO * INS];    // 84 x 84 padded input tile
    __shared__ float s_tmp[HALO * TMPS];  // 84 x 64 horizontally-filtered tile
    __shared__ float s_w[KS];             // separable 1D taps

    const int tid  = threadIdx.x;
    const int img  = blockIdx.x >> 6;          // 0..47  (n*3 + c)
    const int tIdx = blockIdx.x & 63;          // 8x8 tiles of 64x64
    const int c    = img % 3;
    const int X0   = (tIdx & 7) * TILE;
    const int Y0   = (tIdx >> 3) * TILE;

    const float* imgp = x + (size_t)img * IMG * IMG;

    // ---- 1D taps: row sums of the 21x21 kernel (phi[i] * sum(phi) = phi[i]) ----
    if (tid < KS) {
        const float* wr = W + c * (KS * KS) + tid * KS;
        float s = 0.0f;
        #pragma unroll
        for (int j = 0; j < KS; ++j) s += wr[j];
        s_w[tid] = s;
    }

    // ---- stage 84x84 halo tile (reflect-indexed) into LDS ----
#if USE_ASYNC_LDS
    for (int i = tid; i < HALO * HALO; i += 256) {
        const int r  = i / HALO;
        const int cl = i - r * HALO;
        const int gy = reflect512(Y0 + r  - PADR);
        const int gx = reflect512(X0 + cl - PADR);
        // direct global -> LDS copy, no VGPR round trip (ASYNCcnt tracked)
        __builtin_amdgcn_global_load_async_to_lds_b32(
            (as1_int*)(imgp + gy * IMG + gx),
            (as3_int*)&s_in[r * INS + cl],
            0, 0);
    }
    __builtin_amdgcn_s_wait_asynccnt(0);
#else
    for (int i = tid; i < HALO * HALO; i += 256) {
        const int r  = i / HALO;
        const int cl = i - r * HALO;
        const int gy = reflect512(Y0 + r  - PADR);
        const int gx = reflect512(X0 + cl - PADR);
        s_in[r * INS + cl] = imgp[gy * IMG + gx];
    }
#endif
    __syncthreads();

    const int wave = tid >> 5;
    const int lane = tid & 31;
    const int half = lane >> 4;   // 0: lanes 0-15, 1: lanes 16-31
    const int lm   = lane & 15;

    // ---- banded Toeplitz weight fragments (shared by both passes) ----
    // element at K-index k = 4*kc + 2*half + v, band index k - lm -> w[k-lm]
    v2f wfrag[9];
    #pragma unroll
    for (int kc = 0; kc < 9; ++kc) {
        const int k0 = kc * 4 + 2 * half;
        const int i0 = k0 - lm;
        const int i1 = k0 + 1 - lm;
        const float f0 = (i0 >= 0 && i0 < KS) ? s_w[i0] : 0.0f;
        const float f1 = (i1 >= 0 && i1 < KS) ? s_w[i1] : 0.0f;
        wfrag[kc].x = f0;
        wfrag[kc].y = f1;
    }

    // ---- pass 1: horizontal. 24 tiles (6 row-blocks x 4 col-blocks) over 8 waves ----
    // tmp[ry+i][cx+j] = sum_u s_in[ry+i][cx+u] * w[u-j]
    for (int tile = wave; tile < 24; tile += 8) {
        const int t  = tile >> 2;
        const int cx = (tile & 3) << 4;
        const int ry = (t < 5) ? (t << 4) : 68;   // last block overlaps: rows 68..83

        v8f acc = {};
        const float* rowp = &s_in[(ry + lm) * INS + cx + 2 * half];
        #pragma unroll
        for (int kc = 0; kc < 9; ++kc) {
            v2f a;
            a.x = rowp[kc * 4 + 0];
            a.y = rowp[kc * 4 + 1];
            acc = __builtin_amdgcn_wmma_f32_16x16x4_f32(
                false, a, false, wfrag[kc], (short)0, acc, false, false);
        }
        float* tp = &s_tmp[(ry + 8 * half) * TMPS + cx + lm];
        #pragma unroll
        for (int r = 0; r < 8; ++r) tp[r * TMPS] = acc[r];
    }
    __syncthreads();

    // ---- pass 2: vertical. 16 output tiles over 8 waves ----
    // out[y0+i][x0+j] = sum_u w[u-i] * tmp[y0+u][x0+j]
    for (int tile = wave; tile < 16; tile += 8) {
        const int y0 = (tile >> 2) << 4;
        const int x0 = (tile & 3) << 4;

        v8f acc = {};
        const float* colp = &s_tmp[(y0 + 2 * half) * TMPS + x0 + lm];
        #pragma unroll
        for (int kc = 0; kc < 9; ++kc) {
            v2f b;
            b.x = colp[(kc * 4 + 0) * TMPS];
            b.y = colp[(kc * 4 + 1) * TMPS];
            acc = __builtin_amdgcn_wmma_f32_16x16x4_f32(
                false, wfrag[kc], false, b, (short)0, acc, false, false);
        }
        float* op = out + ((size_t)img * IMG + (Y0 + y0 + 8 * half)) * IMG
                        + X0 + x0 + lm;
        #pragma unroll
        for (int r = 0; r < 8; ++r) op[(size_t)r * IMG] = acc[r];
    }
}

extern "C" void kernel_launch(void* const* d_in, const int* in_sizes, int n_in,
                              void* d_out, int out_size, void* d_ws, size_t ws_size,
                              hipStream_t stream) {
    (void)in_sizes; (void)n_in; (void)d_ws; (void)ws_size; (void)out_size;
    const float* x = (const float*)d_in[0];
    const float* W = (const float*)d_in[1];
    float* out = (float*)d_out;

    dim3 grid(48 * 64);   // 16*3 images, 8x8 tiles of 64x64 each
    dim3 block(256);      // 8 waves (wave32)
    hipLaunchKernelGGL(gauss21_sep_wmma, grid, block, 0, stream, x, W, out);
}